// SimpleSSM_12575664242953
// MI455X (gfx1250) — compile-verified
//
#include <hip/hip_runtime.h>
#include <hip/hip_bf16.h>

typedef __attribute__((ext_vector_type(2))) float v2f;
typedef __attribute__((ext_vector_type(8))) float v8f;
typedef int v4i __attribute__((vector_size(16)));

#define BD 32      // batch
#define TD 4096    // time
#define DD 128     // feature dim
#define CH 64      // chunk length (rows per GEMM block)
#define NCH (TD / CH)   // 64 chunks
#define LDP 132    // padded LDS row stride (floats)

// workspace layout (float offsets)
#define OFF_A      0                       // 128x128 combined g-weight (P gW2 gW1)
#define OFF_WO     (OFF_A + 16384)         // 128x128 combined f-weight (fW2 fW1 P^T)
#define OFF_BU     (OFF_WO + 16384)        // 128
#define OFF_BO     (OFF_BU + 128)          // 128
#define OFF_DECAY  (OFF_BO + 128)          // 128  a = 1 - relu(w)
#define OFF_AL     (OFF_DECAY + 128)       // 128  a^CH
#define OFF_T1     (OFF_AL + 128)          // 16384 tmp gW2*gW1
#define OFF_T2     (OFF_T1 + 16384)        // 16384 tmp fW2*fW1
#define OFF_TMPB   (OFF_T2 + 16384)        // 128
#define OFF_E      (OFF_TMPB + 128)        // B*NCH*D chunk-end locals
#define OFF_CARRY  (OFF_E + BD * NCH * DD) // B*NCH*D carries

// ---------------------------------------------------------------------------
// gfx1250 async global->LDS DMA (ASYNCcnt path), guarded so missing builtins
// fall back to the synchronous float4 copy.
#if defined(__has_builtin)
#if __has_builtin(__builtin_amdgcn_global_load_async_to_lds_b128) && \
    __has_builtin(__builtin_amdgcn_s_wait_asynccnt)
#define USE_ASYNC_LDS 1
#endif
#endif

__device__ __forceinline__ void stage16(const float* g, float* l) {
#ifdef USE_ASYNC_LDS
  __builtin_amdgcn_global_load_async_to_lds_b128(
      (__attribute__((address_space(1))) v4i*)g,
      (__attribute__((address_space(3))) v4i*)l, 0, 0);
#else
  *(float4*)l = *(const float4*)g;
#endif
}

__device__ __forceinline__ void stage_wait() {
#ifdef USE_ASYNC_LDS
  __builtin_amdgcn_s_wait_asynccnt(0);
#endif
}

// ---------------------------------------------------------------------------
// Precompute combined weights / biases / decay (one 256-thread block).
//   A  = P @ gW2 @ gW1           u   = x @ A^T + bu
//   Wo = (fW2 @ fW1) @ P^T       out = y @ Wo^T + bo
__global__ __launch_bounds__(256) void ssm_precompute(
    const float* __restrict__ w,  const float* __restrict__ P,
    const float* __restrict__ gW1, const float* __restrict__ gb1,
    const float* __restrict__ gW2, const float* __restrict__ gb2,
    const float* __restrict__ fW1, const float* __restrict__ fb1,
    const float* __restrict__ fW2, const float* __restrict__ fb2,
    float* __restrict__ ws)
{
  const int tid = threadIdx.x;
  for (int idx = tid; idx < 16384; idx += 256) {
    const int i = idx >> 7, j = idx & 127;
    float s1 = 0.f, s2 = 0.f;
    for (int k = 0; k < 128; ++k) {
      s1 += gW2[i * 128 + k] * gW1[k * 128 + j];
      s2 += fW2[i * 128 + k] * fW1[k * 128 + j];
    }
    ws[OFF_T1 + idx] = s1;
    ws[OFF_T2 + idx] = s2;
  }
  if (tid < 128) {
    float sb = gb2[tid], so = fb2[tid];
    for (int k = 0; k < 128; ++k) {
      sb += gb1[k] * gW2[tid * 128 + k];
      so += fb1[k] * fW2[tid * 128 + k];
    }
    ws[OFF_TMPB + tid] = sb;
    ws[OFF_BO + tid]   = so;
    const float ad = 1.f - fmaxf(w[tid], 0.f);
    ws[OFF_DECAY + tid] = ad;
    float p = ad;                       // ad^64 by 6 squarings
    for (int q = 0; q < 6; ++q) p *= p;
    ws[OFF_AL + tid] = p;
  }
  __syncthreads();
  for (int idx = tid; idx < 16384; idx += 256) {
    const int i = idx >> 7, j = idx & 127;
    float sA = 0.f, sW = 0.f;
    for (int k = 0; k < 128; ++k) {
      sA += P[i * 128 + k] * ws[OFF_T1 + k * 128 + j];
      sW += ws[OFF_T2 + i * 128 + k] * P[j * 128 + k];
    }
    ws[OFF_A + idx]  = sA;
    ws[OFF_WO + idx] = sW;
  }
  if (tid < 128) {
    float s = 0.f;
    for (int i = 0; i < 128; ++i) s += ws[OFF_TMPB + i] * P[tid * 128 + i];
    ws[OFF_BU + tid] = s;
  }
}

// ---------------------------------------------------------------------------
// WMMA inner tile: acc[8] (16x128 out tile) += Xs[16x128] * Wlds^T
// f32 16x16x4; A frag: lane(0-15)=M, vgpr v @ half h -> K = k0+2h+v.
// B frag: lane&15 = N, vgpr v @ half h -> K = k0+2h+v. C/D: M = r+8h, N=lane&15.
__device__ __forceinline__ void wmma_tile_f32(
    const float* __restrict__ xs, const float* __restrict__ wlds,
    v8f acc[8], int m0, int lane)
{
  const int half = lane >> 4;
  const int nl   = lane & 15;
  const int arow = (m0 + nl) * LDP;
  for (int k0 = 0; k0 < 128; k0 += 4) {
    v2f af;
    af.x = xs[arow + k0 + 2 * half + 0];
    af.y = xs[arow + k0 + 2 * half + 1];
#pragma unroll
    for (int nt = 0; nt < 8; ++nt) {
      v2f bf;
      const int n = nt * 16 + nl;
      bf.x = wlds[n * LDP + k0 + 2 * half + 0];
      bf.y = wlds[n * LDP + k0 + 2 * half + 1];
      acc[nt] = __builtin_amdgcn_wmma_f32_16x16x4_f32(
          false, af, false, bf, (short)0, acc[nt], false, false);
    }
  }
}

// ---------------------------------------------------------------------------
// Kernel 1: u = x @ A^T + bu for one (batch, chunk) 64-row tile, then local
// first-order scan along t within the chunk (carry-in 0). Writes local y to
// yloc (= d_out) and chunk-end value to e.
__global__ __launch_bounds__(128) void ssm_gemm_g_scan(
    const float* __restrict__ x, const float* __restrict__ ws,
    float* __restrict__ yloc, float* __restrict__ e)
{
  __shared__ __align__(16) float xs[CH * LDP];      // x tile, reused for u
  __shared__ __align__(16) float wlds[DD * LDP];    // combined weight A

  const int tid   = threadIdx.x;
  const int b     = blockIdx.x >> 6;
  const int chunk = blockIdx.x & (NCH - 1);
  const int base  = (b * TD + chunk * CH) * DD;
  const float* __restrict__ A  = ws + OFF_A;
  const float* __restrict__ bu = ws + OFF_BU;
  const float* __restrict__ av = ws + OFF_DECAY;

  // stage x tile + weight (async DMA into LDS when available)
  for (int i = tid; i < CH * DD / 4; i += 128) {
    const int fi = i * 4;
    stage16(x + base + fi, &xs[(fi >> 7) * LDP + (fi & 127)]);
  }
  for (int i = tid; i < DD * DD / 4; i += 128) {
    const int fi = i * 4;
    stage16(A + fi, &wlds[(fi >> 7) * LDP + (fi & 127)]);
  }
  stage_wait();
  __syncthreads();

  const int wv = tid >> 5, lane = tid & 31;
  const int m0 = wv * 16;
  v8f acc[8];
#pragma unroll
  for (int i = 0; i < 8; ++i)
#pragma unroll
    for (int j = 0; j < 8; ++j) acc[i][j] = 0.f;

  wmma_tile_f32(xs, wlds, acc, m0, lane);
  __syncthreads();

  // write u + bias back into xs (reuse)
  {
    const int half = lane >> 4, nl = lane & 15;
#pragma unroll
    for (int nt = 0; nt < 8; ++nt) {
      const int n = nt * 16 + nl;
      const float bn = bu[n];
#pragma unroll
      for (int r = 0; r < 8; ++r)
        xs[(m0 + r + 8 * half) * LDP + n] = acc[nt][r] + bn;
    }
  }
  __syncthreads();

  // local scan: thread tid handles channel d = tid (blockDim == 128)
  {
    const int d = tid;
    const float ad = av[d];
    float y = 0.f;
    for (int j = 0; j < CH; ++j) {
      y = ad * y + xs[j * LDP + d];
      yloc[base + j * DD + d] = y;
    }
    e[(b * NCH + chunk) * DD + tid] = y;
  }
}

// ---------------------------------------------------------------------------
// Kernel 2: per-channel carry propagation across chunks.
__global__ __launch_bounds__(128) void ssm_carry(
    const float* __restrict__ e, const float* __restrict__ aL,
    float* __restrict__ carry)
{
  const int idx = blockIdx.x * 128 + threadIdx.x;   // 0..4095
  const int b = idx >> 7, d = idx & 127;
  const float al = aL[d];
  float c = 0.f;
  for (int ch = 0; ch < NCH; ++ch) {
    carry[(b * NCH + ch) * DD + d] = c;
    c = al * c + e[(b * NCH + ch) * DD + d];
  }
}

// ---------------------------------------------------------------------------
// Kernel 3: DMA raw yloc tile + weight into LDS (overlapped with building the
// a^{j+1} table and carry vector), correct y in-place in LDS, then
// out = y @ Wo^T + bo. In-place on d_out (tile fully read before written).
__global__ __launch_bounds__(128) void ssm_gemm_f_out(
    float* __restrict__ io, const float* __restrict__ ws,
    const float* __restrict__ carry)
{
  __shared__ __align__(16) float xs[CH * LDP];
  __shared__ __align__(16) float wlds[DD * LDP];
  __shared__ __align__(16) float pw[CH * DD];   // a^{j+1} table
  __shared__ __align__(16) float carr[DD];

  const int tid   = threadIdx.x;
  const int b     = blockIdx.x >> 6;
  const int chunk = blockIdx.x & (NCH - 1);
  const int base  = (b * TD + chunk * CH) * DD;
  const float* __restrict__ Wo = ws + OFF_WO;
  const float* __restrict__ bo = ws + OFF_BO;
  const float* __restrict__ av = ws + OFF_DECAY;

  // kick off DMA of raw y tile + weight
  for (int i = tid; i < CH * DD / 4; i += 128) {
    const int fi = i * 4;
    stage16(io + base + fi, &xs[(fi >> 7) * LDP + (fi & 127)]);
  }
  for (int i = tid; i < DD * DD / 4; i += 128) {
    const int fi = i * 4;
    stage16(Wo + fi, &wlds[(fi >> 7) * LDP + (fi & 127)]);
  }

  // overlap: per-channel carry + power table (VALU while DMA in flight)
  {
    const int d = tid;                           // blockDim == 128
    carr[d] = carry[(b * NCH + chunk) * DD + d];
    const float ad = av[d];
    float p = 1.f;
    for (int j = 0; j < CH; ++j) { p *= ad; pw[j * DD + d] = p; }
  }
  stage_wait();
  __syncthreads();

  // in-place carry correction on the LDS tile: y += a^{j+1} * carry
  for (int i = tid; i < CH * DD; i += 128) {
    const int r = i >> 7, c = i & 127;
    xs[r * LDP + c] += pw[i] * carr[c];
  }
  __syncthreads();

  const int wv = tid >> 5, lane = tid & 31;
  const int m0 = wv * 16;
  v8f acc[8];
#pragma unroll
  for (int i = 0; i < 8; ++i)
#pragma unroll
    for (int j = 0; j < 8; ++j) acc[i][j] = 0.f;

  wmma_tile_f32(xs, wlds, acc, m0, lane);

  // write final output (+ bias) straight from accumulators
  {
    const int half = lane >> 4, nl = lane & 15;
#pragma unroll
    for (int nt = 0; nt < 8; ++nt) {
      const int n = nt * 16 + nl;
      const float bn = bo[n];
#pragma unroll
      for (int r = 0; r < 8; ++r)
        io[base + (m0 + r + 8 * half) * DD + n] = acc[nt][r] + bn;
    }
  }
}

// ---------------------------------------------------------------------------
extern "C" void kernel_launch(void* const* d_in, const int* in_sizes, int n_in,
                              void* d_out, int out_size, void* d_ws, size_t ws_size,
                              hipStream_t stream) {
  const float* x   = (const float*)d_in[0];
  const float* w   = (const float*)d_in[1];
  const float* P   = (const float*)d_in[2];
  const float* gW1 = (const float*)d_in[3];
  const float* gb1 = (const float*)d_in[4];
  const float* gW2 = (const float*)d_in[5];
  const float* gb2 = (const float*)d_in[6];
  const float* fW1 = (const float*)d_in[7];
  const float* fb1 = (const float*)d_in[8];
  const float* fW2 = (const float*)d_in[9];
  const float* fb2 = (const float*)d_in[10];
  float* out = (float*)d_out;
  float* ws  = (float*)d_ws;

  ssm_precompute<<<1, 256, 0, stream>>>(w, P, gW1, gb1, gW2, gb2,
                                        fW1, fb1, fW2, fb2, ws);
  const int nblocks = BD * NCH;  // 2048
  ssm_gemm_g_scan<<<nblocks, 128, 0, stream>>>(x, ws, out, ws + OFF_E);
  ssm_carry<<<BD, 128, 0, stream>>>(ws + OFF_E, ws + OFF_AL, ws + OFF_CARRY);
  ssm_gemm_f_out<<<nblocks, 128, 0, stream>>>(out, ws, ws + OFF_CARRY);
}